// MonarchBlockDiagonalMHA_35828617183993
// MI455X (gfx1250) — compile-verified
//
#include <hip/hip_runtime.h>

// ---------------------------------------------------------------------------
// Monarch block-diagonal attention on gfx1250 (MI455X), fp32 WMMA throughout.
// b=4, h=12, S=2048, D=64, B_SIZE=64, nb=32, NUM_STEPS=2, STEP_SIZE=0.5
// ---------------------------------------------------------------------------

#define S_   2048
#define D_   64
#define BS_  64
#define NB_  32
#define CW_  96          // B_SIZE + nb
#define NH_  48          // b*h heads

typedef __attribute__((ext_vector_type(2))) float v2f;
typedef __attribute__((ext_vector_type(8))) float v8f;

// ---- workspace layout (floats) --------------------------------------------
static constexpr size_t SZ_BL   = (size_t)NH_ * S_ * CW_;        // bl_params
static constexpr size_t SZ_R    = (size_t)NH_ * NB_ * BS_ * BS_; // r_params
static constexpr size_t O_BL    = 0;
static constexpr size_t O_R     = O_BL  + SZ_BL;
static constexpr size_t O_DBL   = O_R   + SZ_R;    // raw d_bl (S x 96 per head)
static constexpr size_t O_T1    = O_DBL + SZ_BL;   // t1[k][j][d]; reused as Y
static constexpr size_t O_T2    = O_T1  + SZ_R;    // t2[k][j][d]
static constexpr size_t O_INVBL = O_T2  + SZ_R;    // per-row 1/||bl_row|| (S/head)
static constexpr size_t O_INVR  = O_INVBL + (size_t)NH_ * S_;
static constexpr size_t O_R2S   = O_INVR  + (size_t)NH_ * S_;    // r2sum[k*64+j]
static constexpr size_t O_DAD   = O_R2S   + (size_t)NH_ * S_;    // diagadd[k*64+j]
static constexpr size_t O_L2S   = O_DAD   + (size_t)NH_ * S_;    // l2sum[j*32+k]
// total floats = O_L2S + NH_*S_  (~38.3M floats = ~153 MB)

// ---- WMMA helpers ----------------------------------------------------------
__device__ __forceinline__ int frag_half() { return (threadIdx.x & 31) >> 4; }
__device__ __forceinline__ int frag_l15()  { return threadIdx.x & 15; }
__device__ __forceinline__ int frag_m(int r) { return r + frag_half() * 8; }

// acc += sgn * A(16xK) * B(Kx16).  AT: A stored transposed (KxM row-major).
// BT: B stored transposed (NxK row-major).  lda/ldb are row strides.
template <bool AT, bool BT>
__device__ __forceinline__ v8f wmma_tile(const float* A, int lda, float asgn,
                                         const float* B, int ldb, int K, v8f acc) {
  const int half = frag_half();
  const int l15  = frag_l15();
  for (int k0 = 0; k0 < K; k0 += 4) {
    const int ka = k0 + half * 2;
    v2f a, b;
    if (AT) { a.x = asgn * A[ka * lda + l15];  a.y = asgn * A[(ka + 1) * lda + l15]; }
    else    { a.x = asgn * A[l15 * lda + ka];  a.y = asgn * A[l15 * lda + ka + 1]; }
    if (BT) { b.x = B[l15 * ldb + ka];         b.y = B[l15 * ldb + ka + 1]; }
    else    { b.x = B[ka * ldb + l15];         b.y = B[(ka + 1) * ldb + l15]; }
    acc = __builtin_amdgcn_wmma_f32_16x16x4_f32(false, a, false, b, (short)0, acc,
                                                false, false);
  }
  return acc;
}

// ---- kernel 0: init parameters --------------------------------------------
__global__ __launch_bounds__(256) void k_init(float* ws) {
  const size_t tid = (size_t)blockIdx.x * 256 + threadIdx.x;
  if (tid < SZ_BL) ws[O_BL + tid] = 0.1020620726159658f;  // 1/sqrt(96)
  if (tid < SZ_R)  ws[O_R + tid]  = 0.125f;               // 1/sqrt(64)
}

// ---- kernel 1: per-row inverse norms --------------------------------------
__global__ __launch_bounds__(256) void k_norms(float* ws) {
  const size_t tid = (size_t)blockIdx.x * 256 + threadIdx.x;
  const size_t NR = (size_t)NH_ * S_;
  if (tid < NR) {
    const float* row = ws + O_BL + tid * CW_;
    float s = 0.f;
    for (int c = 0; c < CW_; ++c) s += row[c] * row[c];
    ws[O_INVBL + tid] = (s > 0.f) ? 1.0f / sqrtf(s) : 0.f;
  } else if (tid < 2 * NR) {
    const size_t r = tid - NR;
    const float* row = ws + O_R + r * BS_;
    float s = 0.f;
    for (int c = 0; c < BS_; ++c) s += row[c] * row[c];
    ws[O_INVR + r] = (s > 0.f) ? 1.0f / sqrtf(s) : 0.f;
  }
}

// ---- kernel 2: per-(head,block) grads: d_bd, t1, r2sum, diagadd -----------
__global__ __launch_bounds__(256) void k_grad_block(const float* __restrict__ q,
                                                    const float* __restrict__ kk,
                                                    float* ws) {
  __shared__ float Qs[BS_ * D_];   // Q block, scaled
  __shared__ float Ks[BS_ * D_];   // K block
  __shared__ float Rsq[BS_ * BS_]; // right = r_s^2
  __shared__ float DiagL[BS_];

  const int blk = blockIdx.x;
  const int hb = blk >> 5, kb = blk & 31;
  const float* qh = q  + (size_t)hb * S_ * D_ + (size_t)kb * BS_ * D_;
  const float* kh = kk + (size_t)hb * S_ * D_ + (size_t)kb * BS_ * D_;
  float* blp   = ws + O_BL    + (size_t)hb * S_ * CW_;
  float* rp    = ws + O_R     + (size_t)hb * NB_ * BS_ * BS_ + (size_t)kb * BS_ * BS_;
  float* invbl = ws + O_INVBL + (size_t)hb * S_;
  float* invr  = ws + O_INVR  + (size_t)hb * S_;
  float* dbl   = ws + O_DBL   + (size_t)hb * S_ * CW_;
  float* t1    = ws + O_T1    + (size_t)hb * NB_ * BS_ * BS_ + (size_t)kb * BS_ * BS_;
  float* r2s   = ws + O_R2S   + (size_t)hb * S_;
  float* dad   = ws + O_DAD   + (size_t)hb * S_;

  const int tid = threadIdx.x;
  for (int i = tid; i < BS_ * D_; i += 256) {
    Qs[i] = qh[i] * 0.125f;
    Ks[i] = kh[i];
    float rs = rp[i] * invr[kb * 64 + (i >> 6)];
    Rsq[i] = rs * rs;
  }
  if (tid < BS_) {
    const int row = kb * 64 + tid;
    float v = blp[row * CW_ + 64 + kb] * invbl[row];
    DiagL[tid] = v * v;
  }
  __syncthreads();

  if (tid < BS_) {  // r2sum / diagadd row reductions
    const int j = tid, row = kb * 64 + j;
    const float inv = invbl[row];
    float s2 = 0.f, sd = 0.f;
    for (int i = 0; i < BS_; ++i) {
      float r = Rsq[j * 64 + i];
      float b = blp[row * CW_ + i] * inv;
      s2 += r * r;
      sd += r * b * b;
    }
    r2s[row] = s2;
    dad[row] = sd;
  }

  const int wid = tid >> 5;
  for (int t = wid; t < 32; t += 8) {
    const int which = t >> 4;        // 0: d_bd tiles, 1: t1 tiles
    const int tt = t & 15;
    const int tj = tt >> 2, ti = tt & 3;
    if (which == 0) {
      v8f acc;
      #pragma unroll
      for (int r = 0; r < 8; ++r) {
        const int m = tj * 16 + frag_m(r), n = ti * 16 + frag_l15();
        const int row = kb * 64 + m;
        float b = blp[row * CW_ + n] * invbl[row];
        acc[r] = b * b + DiagL[m] * Rsq[m * 64 + n];
      }
      // d_bd = BD + diag_l*right - Q@K^T  (negate A to subtract)
      acc = wmma_tile<false, true>(Qs + tj * 16 * 64, 64, -1.f,
                                   Ks + ti * 16 * 64, 64, 64, acc);
      #pragma unroll
      for (int r = 0; r < 8; ++r) {
        const int m = tj * 16 + frag_m(r), n = ti * 16 + frag_l15();
        dbl[(kb * 64 + m) * CW_ + n] = acc[r];
      }
    } else {
      v8f acc = 0.0f;  // t1 = right @ K
      acc = wmma_tile<false, false>(Rsq + tj * 16 * 64, 64, 1.f,
                                    Ks + ti * 16, 64, 64, acc);
      #pragma unroll
      for (int r = 0; r < 8; ++r) {
        const int m = tj * 16 + frag_m(r), n = ti * 16 + frag_l15();
        t1[m * 64 + n] = acc[r];
      }
    }
  }
}

// ---- kernel 3: per-(head,j) grads: d_left, l2sum, t2 ----------------------
__global__ __launch_bounds__(256) void k_grad_j(const float* __restrict__ q,
                                                float* ws) {
  __shared__ float Lj[NB_ * NB_];   // left[j, l, k]
  __shared__ float Qj[NB_ * D_];    // q[l, j, :]
  __shared__ float T1j[NB_ * D_];   // t1[k, j, :]

  const int blk = blockIdx.x;
  const int hb = blk >> 6, j = blk & 63;
  const float* qh = q + (size_t)hb * S_ * D_;
  float* blp   = ws + O_BL    + (size_t)hb * S_ * CW_;
  float* invbl = ws + O_INVBL + (size_t)hb * S_;
  float* dbl   = ws + O_DBL   + (size_t)hb * S_ * CW_;
  float* t1h   = ws + O_T1    + (size_t)hb * NB_ * BS_ * BS_;
  float* t2h   = ws + O_T2    + (size_t)hb * NB_ * BS_ * BS_;
  float* r2s   = ws + O_R2S   + (size_t)hb * S_;
  float* dad   = ws + O_DAD   + (size_t)hb * S_;
  float* l2s   = ws + O_L2S   + (size_t)hb * S_;

  const int tid = threadIdx.x;
  for (int i = tid; i < NB_ * NB_; i += 256) {
    const int l = i >> 5, k2 = i & 31;
    float v = blp[(l * 64 + j) * CW_ + 64 + k2] * invbl[l * 64 + j];
    Lj[i] = v * v;
  }
  for (int i = tid; i < NB_ * D_; i += 256) {
    const int l = i >> 6, d = i & 63;
    Qj[i]  = qh[(l * 64 + j) * 64 + d] * 0.125f;
    T1j[i] = t1h[l * (BS_ * BS_) + j * 64 + d];
  }
  __syncthreads();

  if (tid < NB_) {  // l2sum[j,k]
    float s = 0.f;
    for (int l = 0; l < NB_; ++l) { float v = Lj[l * 32 + tid]; s += v * v; }
    l2s[j * 32 + tid] = s;
  }

  const int wid = tid >> 5;
  if (wid < 4) {  // M = Qj @ T1j^T ; d_left = left*r2sum - M (+diag)
    const int tj = wid >> 1, ti = wid & 1;
    v8f acc = 0.0f;
    acc = wmma_tile<false, true>(Qj + tj * 16 * 64, 64, 1.f,
                                 T1j + ti * 16 * 64, 64, 64, acc);
    #pragma unroll
    for (int r = 0; r < 8; ++r) {
      const int l = tj * 16 + frag_m(r), k2 = ti * 16 + frag_l15();
      float v = Lj[l * 32 + k2] * r2s[k2 * 64 + j] - acc[r];
      if (l == k2) v += dad[l * 64 + j];
      dbl[(l * 64 + j) * CW_ + 64 + k2] = v;
    }
  }
  {  // t2[j,k,d] = left^T @ Qj  (contract over l, K=32)
    const int tj = wid >> 2, ti = wid & 3;
    v8f acc = 0.0f;
    acc = wmma_tile<true, false>(Lj + tj * 16, 32, 1.f,
                                 Qj + ti * 16, 64, 32, acc);
    #pragma unroll
    for (int r = 0; r < 8; ++r) {
      const int k2 = tj * 16 + frag_m(r), d = ti * 16 + frag_l15();
      t2h[k2 * (BS_ * BS_) + j * 64 + d] = acc[r];
    }
  }
}

// ---- kernel 4: per-(head,block) d_right + in-place r_params update --------
__global__ __launch_bounds__(256) void k_grad_r(const float* __restrict__ kk,
                                                float* ws) {
  __shared__ float T2s[BS_ * D_];
  __shared__ float Ks[BS_ * D_];
  __shared__ float DRs[BS_ * BS_];

  const int blk = blockIdx.x;
  const int hb = blk >> 5, kb = blk & 31;
  const float* kh = kk + (size_t)hb * S_ * D_ + (size_t)kb * BS_ * D_;
  float* blp   = ws + O_BL    + (size_t)hb * S_ * CW_;
  float* rp    = ws + O_R     + (size_t)hb * NB_ * BS_ * BS_ + (size_t)kb * BS_ * BS_;
  float* invbl = ws + O_INVBL + (size_t)hb * S_;
  float* invr  = ws + O_INVR  + (size_t)hb * S_;
  float* t2h   = ws + O_T2    + (size_t)hb * NB_ * BS_ * BS_ + (size_t)kb * BS_ * BS_;
  float* l2s   = ws + O_L2S   + (size_t)hb * S_;

  const int tid = threadIdx.x;
  for (int i = tid; i < BS_ * D_; i += 256) {
    T2s[i] = t2h[i];
    Ks[i]  = kh[i];
  }
  __syncthreads();

  const int wid = tid >> 5;
  for (int t = wid; t < 16; t += 8) {
    const int tj = t >> 2, ti = t & 3;
    v8f acc;
    #pragma unroll
    for (int r = 0; r < 8; ++r) {
      const int m = tj * 16 + frag_m(r), n = ti * 16 + frag_l15();
      const int row = kb * 64 + m;
      const float ib = invbl[row];
      float bd = blp[row * CW_ + n] * ib;         bd *= bd;
      float dl = blp[row * CW_ + 64 + kb] * ib;   dl *= dl;
      float rv = rp[m * 64 + n] * invr[row];      rv *= rv;
      acc[r] = rv * l2s[m * 32 + kb] + dl * bd;
    }
    // d_right -= T2 @ K^T
    acc = wmma_tile<false, true>(T2s + tj * 16 * 64, 64, -1.f,
                                 Ks + ti * 16 * 64, 64, 64, acc);
    #pragma unroll
    for (int r = 0; r < 8; ++r) {
      const int m = tj * 16 + frag_m(r), n = ti * 16 + frag_l15();
      DRs[m * 64 + n] = acc[r];
    }
  }
  __syncthreads();

  if (tid < BS_) {  // tangent projection + SGD update, one row per thread
    const int j = tid, row = kb * 64 + j;
    const float inv = invr[row];
    float dot = 0.f;
    for (int i = 0; i < BS_; ++i) {
      float rs = rp[j * 64 + i] * inv;
      dot += rs * (2.f * DRs[j * 64 + i] * rs);
    }
    for (int i = 0; i < BS_; ++i) {
      float rs = rp[j * 64 + i] * inv;
      float dr = 2.f * DRs[j * 64 + i] * rs;
      float df = (dr - rs * dot) * inv;
      rp[j * 64 + i] -= 0.5f * df;
    }
  }
}

// ---- kernel 5: per-row bl tangent projection + update ---------------------
__global__ __launch_bounds__(256) void k_bl_update(float* ws) {
  const size_t tid = (size_t)blockIdx.x * 256 + threadIdx.x;
  const size_t NR = (size_t)NH_ * S_;
  if (tid >= NR) return;
  float* blrow = ws + O_BL + tid * CW_;
  const float* drow = ws + O_DBL + tid * CW_;
  const float inv = ws[O_INVBL + tid];
  float dot = 0.f;
  for (int c = 0; c < CW_; ++c) {
    float bs = blrow[c] * inv;
    dot += bs * (2.f * drow[c] * bs);
  }
  for (int c = 0; c < CW_; ++c) {
    float bs = blrow[c] * inv;
    float d = 2.f * drow[c] * bs;
    blrow[c] -= 0.5f * ((d - bs * dot) * inv);
  }
}

// ---- kernel 6: Y = right @ X  (stored in t1 buffer) -----------------------
__global__ __launch_bounds__(256) void k_Y(const float* __restrict__ val,
                                           float* ws) {
  __shared__ float Rsq[BS_ * BS_];
  __shared__ float Xs[BS_ * D_];
  const int blk = blockIdx.x;
  const int hb = blk >> 5, kb = blk & 31;
  const float* xh = val + (size_t)hb * S_ * D_ + (size_t)kb * BS_ * D_;
  float* rp   = ws + O_R    + (size_t)hb * NB_ * BS_ * BS_ + (size_t)kb * BS_ * BS_;
  float* invr = ws + O_INVR + (size_t)hb * S_;
  float* yb   = ws + O_T1   + (size_t)hb * NB_ * BS_ * BS_ + (size_t)kb * BS_ * BS_;

  const int tid = threadIdx.x;
  for (int i = tid; i < BS_ * D_; i += 256) {
    float rs = rp[i] * invr[kb * 64 + (i >> 6)];
    Rsq[i] = rs * rs;
    Xs[i]  = xh[i];
  }
  __syncthreads();

  const int wid = tid >> 5;
  for (int t = wid; t < 16; t += 8) {
    const int tj = t >> 2, ti = t & 3;
    v8f acc = 0.0f;
    acc = wmma_tile<false, false>(Rsq + tj * 16 * 64, 64, 1.f,
                                  Xs + ti * 16, 64, 64, acc);
    #pragma unroll
    for (int r = 0; r < 8; ++r) {
      const int m = tj * 16 + frag_m(r), n = ti * 16 + frag_l15();
      yb[m * 64 + n] = acc[r];
    }
  }
}

// ---- kernel 7: Z[l,j,:] = leftj @ Yj  (writes full output) ----------------
__global__ __launch_bounds__(256) void k_zleft(float* __restrict__ out,
                                               float* ws) {
  __shared__ float Lj[NB_ * NB_];
  __shared__ float Yj[NB_ * D_];
  const int blk = blockIdx.x;
  const int hb = blk >> 6, j = blk & 63;
  float* blp   = ws + O_BL    + (size_t)hb * S_ * CW_;
  float* invbl = ws + O_INVBL + (size_t)hb * S_;
  float* yb    = ws + O_T1    + (size_t)hb * NB_ * BS_ * BS_;
  float* oh    = out + (size_t)hb * S_ * D_;

  const int tid = threadIdx.x;
  for (int i = tid; i < NB_ * NB_; i += 256) {
    const int l = i >> 5, k2 = i & 31;
    float v = blp[(l * 64 + j) * CW_ + 64 + k2] * invbl[l * 64 + j];
    Lj[i] = v * v;
  }
  for (int i = tid; i < NB_ * D_; i += 256) {
    const int k2 = i >> 6, d = i & 63;
    Yj[i] = yb[k2 * (BS_ * BS_) + j * 64 + d];
  }
  __syncthreads();

  const int wid = tid >> 5;
  const int tj = wid >> 2, ti = wid & 3;
  v8f acc = 0.0f;
  acc = wmma_tile<false, false>(Lj + tj * 16 * 32, 32, 1.f,
                                Yj + ti * 16, 64, 32, acc);
  #pragma unroll
  for (int r = 0; r < 8; ++r) {
    const int l = tj * 16 + frag_m(r), d = ti * 16 + frag_l15();
    oh[(size_t)(l * 64 + j) * 64 + d] = acc[r];
  }
}

// ---- kernel 8: out += block_diag @ X --------------------------------------
__global__ __launch_bounds__(256) void k_zbd(const float* __restrict__ val,
                                             float* __restrict__ out,
                                             float* ws) {
  __shared__ float BDs[BS_ * BS_];
  __shared__ float Xs[BS_ * D_];
  const int blk = blockIdx.x;
  const int hb = blk >> 5, lb = blk & 31;
  const float* xh = val + (size_t)hb * S_ * D_ + (size_t)lb * BS_ * D_;
  float* blp   = ws + O_BL    + (size_t)hb * S_ * CW_;
  float* invbl = ws + O_INVBL + (size_t)hb * S_;
  float* oh    = out + (size_t)hb * S_ * D_ + (size_t)lb * BS_ * D_;

  const int tid = threadIdx.x;
  for (int i = tid; i < BS_ * D_; i += 256) {
    const int j = i >> 6, c = i & 63;
    float v = blp[(lb * 64 + j) * CW_ + c] * invbl[lb * 64 + j];
    BDs[i] = v * v;
    Xs[i]  = xh[i];
  }
  __syncthreads();

  const int wid = tid >> 5;
  for (int t = wid; t < 16; t += 8) {
    const int tj = t >> 2, ti = t & 3;
    v8f acc = 0.0f;
    acc = wmma_tile<false, false>(BDs + tj * 16 * 64, 64, 1.f,
                                  Xs + ti * 16, 64, 64, acc);
    #pragma unroll
    for (int r = 0; r < 8; ++r) {
      const int m = tj * 16 + frag_m(r), n = ti * 16 + frag_l15();
      oh[m * 64 + n] += acc[r];
    }
  }
}

// ---------------------------------------------------------------------------
extern "C" void kernel_launch(void* const* d_in, const int* in_sizes, int n_in,
                              void* d_out, int out_size, void* d_ws, size_t ws_size,
                              hipStream_t stream) {
  const float* q = (const float*)d_in[0];
  const float* k = (const float*)d_in[1];
  const float* v = (const float*)d_in[2];
  float* out = (float*)d_out;
  float* ws = (float*)d_ws;

  const int initN  = (int)((SZ_BL + 255) / 256);
  const int normN  = (int)((2 * (size_t)NH_ * S_ + 255) / 256);
  const int blupN  = (int)(((size_t)NH_ * S_ + 255) / 256);
  const int blkG   = NH_ * NB_;   // 1536 workgroups
  const int jG     = NH_ * BS_;   // 3072 workgroups

  k_init<<<initN, 256, 0, stream>>>(ws);

  for (int step = 0; step < 2; ++step) {
    k_norms<<<normN, 256, 0, stream>>>(ws);
    k_grad_block<<<blkG, 256, 0, stream>>>(q, k, ws);
    k_grad_j<<<jG, 256, 0, stream>>>(q, ws);
    k_grad_r<<<blkG, 256, 0, stream>>>(k, ws);
    k_bl_update<<<blupN, 256, 0, stream>>>(ws);
  }

  k_norms<<<normN, 256, 0, stream>>>(ws);
  k_Y<<<blkG, 256, 0, stream>>>(v, ws);
  k_zleft<<<jG, 256, 0, stream>>>(out, ws);
  k_zbd<<<blkG, 256, 0, stream>>>(v, out, ws);
}